// WaveletTransform3D_28097676051225
// MI455X (gfx1250) — compile-verified
//
#include <hip/hip_runtime.h>

// 3D Haar wavelet (4,1,9,512,512) f32 -> LLL (4,1,8,512,512) + 7 high bands (4,56,512,512).
// Memory-bound: 256MB written, 36MB read (input is L2-resident; L2 = 192MB).
// Strategy: async global->LDS tile staging (gfx1250 GLOBAL_LOAD_ASYNC_TO_LDS_B128,
// S_WAIT_ASYNCCNT), b128 vector accesses, non-temporal b128 stores for write-once output.

typedef float v4f __attribute__((ext_vector_type(4)));

#define HAAR_S 0.35355339059327373f   // 1/(2*sqrt(2))

constexpr int FIN  = 9;    // input frames
constexpr int FO   = 8;    // output frames
constexpr int H    = 512;
constexpr int W    = 512;
constexpr int TH   = 8;    // output rows per block
constexpr int ROWP = 516;  // padded LDS row stride in floats (2064B, 16B multiple)

__global__ __launch_bounds__(256) void haar3d_kernel(const float* __restrict__ x,
                                                     float* __restrict__ out,
                                                     int B) {
    // LDS tile: 2 frames x 9 rows (h-1 .. h+7, clamped) x 512 floats (padded row)
    __shared__ __attribute__((aligned(16))) float tile[2 * 9 * ROWP];

    const int bid   = blockIdx.x;         // B * 8 * 64 blocks
    const int tileh = bid & 63;           // 64 h-tiles of 8 rows
    const int f     = (bid >> 6) & 7;     // output frame
    const int b     = bid >> 9;           // batch
    const int h0    = tileh * TH;
    const int tid   = threadIdx.x;

    // ---- Stage input tile into LDS with async b128 loads ---------------------
    // 2 frames * 9 rows * 128 float4 units = 2304 units; 9 per thread.
    const float* xb = x + (size_t)(b * FIN + f) * H * W;
#pragma unroll
    for (int i = 0; i < 9; ++i) {
        int u   = tid + i * 256;
        int fi  = u / 1152;          // 0..1 frame
        int rem = u - fi * 1152;
        int r   = rem >> 7;          // 0..8 staged row
        int w4  = rem & 127;         // float4 index in row
        int gh  = h0 - 1 + r;        // replicate-clamp top edge only
        if (gh < 0) gh = 0;
        const float* gp = xb + (size_t)fi * (H * W) + gh * W + w4 * 4;
        unsigned lp = (unsigned)(unsigned long long)(const void*)
                      &tile[(fi * 9 + r) * ROWP + w4 * 4];
        unsigned long long ga = (unsigned long long)(const void*)gp;
        asm volatile("global_load_async_to_lds_b128 %0, %1, off"
                     :: "v"(lp), "v"(ga) : "memory");
    }
    asm volatile("s_wait_asynccnt 0" ::: "memory");
    __syncthreads();

    // ---- Compute: one wave per output row -----------------------------------
    const int ty = tid >> 5;   // 0..7 -> output row h0+ty
    const int tx = tid & 31;
    const int h  = h0 + ty;

    const float* r0f0 = &tile[(0 * 9 + ty    ) * ROWP];  // frame f,   row h-1
    const float* r1f0 = &tile[(0 * 9 + ty + 1) * ROWP];  // frame f,   row h
    const float* r0f1 = &tile[(1 * 9 + ty    ) * ROWP];  // frame f+1, row h-1
    const float* r1f1 = &tile[(1 * 9 + ty + 1) * ROWP];  // frame f+1, row h

    float*       outL = out + ((size_t)((b * FO + f) * H + h)) * W;
    float*       outH = out + (size_t)B * FO * H * W;           // high-band base
    const size_t hb   = ((size_t)(b * 56 + f) * H + h) * W;     // band-0 plane offset
    const size_t bs   = (size_t)FO * H * W;                     // per-band stride

    const float s = HAAR_S;

#pragma unroll
    for (int k = 0; k < 4; ++k) {
        int w = tx * 4 + k * 128;

        v4f a0 = *(const v4f*)(r0f0 + w);   // x[f  ][h-1][w..w+3]
        v4f b0 = *(const v4f*)(r1f0 + w);   // x[f  ][h  ]
        v4f a1 = *(const v4f*)(r0f1 + w);   // x[f+1][h-1]
        v4f b1 = *(const v4f*)(r1f1 + w);   // x[f+1][h  ]

        int wm = w ? (w - 1) : 0;           // replicate at w==0
        float a0l = r0f0[wm], b0l = r1f0[wm], a1l = r0f1[wm], b1l = r1f1[wm];

        // temporal (valid): L=(x_f+x_{f+1})*s, H=(x_{f+1}-x_f)*s
        v4f tl0 = (a0 + a1) * s, th0 = (a1 - a0) * s;
        v4f tl1 = (b0 + b1) * s, th1 = (b1 - b0) * s;
        float tl0l = (a0l + a1l) * s, th0l = (a1l - a0l) * s;
        float tl1l = (b0l + b1l) * s, th1l = (b1l - b0l) * s;

        // h axis (replicate-pad): uses rows h-1, h
        v4f ll = (tl0 + tl1) * s, lh = (tl1 - tl0) * s;
        v4f hl = (th0 + th1) * s, hh = (th1 - th0) * s;
        float llw = (tl0l + tl1l) * s, lhw = (tl1l - tl0l) * s;
        float hlw = (th0l + th1l) * s, hhw = (th1l - th0l) * s;

        // w axis (replicate-pad): prev = value at w-1
        v4f llp = {llw, ll.x, ll.y, ll.z};
        v4f lhp = {lhw, lh.x, lh.y, lh.z};
        v4f hlp = {hlw, hl.x, hl.y, hl.z};
        v4f hhp = {hhw, hh.x, hh.y, hh.z};

        v4f LLL = (llp + ll) * s, LLH = (ll - llp) * s;
        v4f LHL = (lhp + lh) * s, LHH = (lh - lhp) * s;
        v4f HLL = (hlp + hl) * s, HLH = (hl - hlp) * s;
        v4f HHL = (hhp + hh) * s, HHH = (hh - hhp) * s;

        // write-once output: non-temporal b128 stores (keep input resident in L2)
        __builtin_nontemporal_store(LLL, (v4f*)(outL + w));
        __builtin_nontemporal_store(LLH, (v4f*)(outH + hb + 0 * bs + w));
        __builtin_nontemporal_store(LHL, (v4f*)(outH + hb + 1 * bs + w));
        __builtin_nontemporal_store(LHH, (v4f*)(outH + hb + 2 * bs + w));
        __builtin_nontemporal_store(HLL, (v4f*)(outH + hb + 3 * bs + w));
        __builtin_nontemporal_store(HLH, (v4f*)(outH + hb + 4 * bs + w));
        __builtin_nontemporal_store(HHL, (v4f*)(outH + hb + 5 * bs + w));
        __builtin_nontemporal_store(HHH, (v4f*)(outH + hb + 6 * bs + w));
    }
}

extern "C" void kernel_launch(void* const* d_in, const int* in_sizes, int n_in,
                              void* d_out, int out_size, void* d_ws, size_t ws_size,
                              hipStream_t stream) {
    (void)n_in; (void)out_size; (void)d_ws; (void)ws_size;
    const float* x = (const float*)d_in[0];
    float* out = (float*)d_out;
    int B = in_sizes[0] / (FIN * H * W);   // = 4 for the reference shapes
    dim3 grid(B * FO * (H / TH));          // 4 * 8 * 64 = 2048 blocks
    haar3d_kernel<<<grid, 256, 0, stream>>>(x, out, B);
}